// RotatedPolygonLoss_61134564491412
// MI455X (gfx1250) — compile-verified
//
#include <hip/hip_runtime.h>
#include <hip/hip_bf16.h>

// Chamfer distance vs rotated polygon templates, fused single pass.
// One workgroup (256 threads = 8 wave32) per batch element b.
//   - TDM tensor_load_to_lds stages examples[idx] (16 KB) into LDS once.
//   - Each V_WMMA_F32_16X16X4_F32 computes a 16x16 block of squared distances
//     via the rank-4 factorization d2 = [gx,gy,|g|^2,1] . [-2yx,-2yy,1,|y|^2],
//     covering 2 rotation candidates (rows) x 8 predicted points (cols).
//   - Packed butterfly reductions (17 shuffles/tile instead of 36) produce
//     mean_i min_j + mean_j min_i per candidate; block-min over 256 candidates.

typedef float v2f __attribute__((ext_vector_type(2)));
typedef float v8f __attribute__((ext_vector_type(8)));
typedef unsigned int u32;
typedef u32 u32x4 __attribute__((ext_vector_type(4)));
typedef int  i32x4 __attribute__((ext_vector_type(4)));
typedef int  i32x8 __attribute__((ext_vector_type(8)));

#define GRAN 256
#define MAXN 8
#define GT_FLOATS (GRAN * MAXN * 2)   // 4096 floats = 16 KB per template

__global__ __launch_bounds__(256) void rotpoly_chamfer_wmma(
    const float* __restrict__ y,        // [B, 8, 2]
    const int*   __restrict__ nv,       // [B] in {3..8}
    const float* __restrict__ labels,   // [6] = {3,4,5,6,7,8}
    const float* __restrict__ ex,       // [6, 256, 8, 2]
    float* __restrict__ out,            // [B]
    int B)
{
  __shared__ float lds_gt[GT_FLOATS];   // staged template, 16 KB
  __shared__ float sbest[8];

  const int b    = blockIdx.x;
  const int lane = threadIdx.x & 31;
  const int wave = threadIdx.x >> 5;
  const int j    = lane & 15;           // column index within 16-wide tile
  const bool hi    = lane >= 16;        // high half of wave32
  const bool valid = j < MAXN;          // only 8 real predicted points

  // template index: the k with labels[k] == n[b] (== n[b]-3 canonically).
  // readfirstlane keeps it scalar so the TDM descriptor stays in SGPRs.
  const int nb = nv[b];
  int idx = 0;
#pragma unroll
  for (int k = 5; k >= 1; --k)
    if (labels[k] == (float)nb) idx = k;
  idx = __builtin_amdgcn_readfirstlane(idx);

  const float* gt = ex + (size_t)idx * GT_FLOATS;

  // ---- Stage the 16 KB template into LDS via the Tensor Data Mover ----
#if __has_builtin(__builtin_amdgcn_tensor_load_to_lds) && \
    __has_builtin(__builtin_amdgcn_s_wait_tensorcnt)
  if (wave == 0) {
    const unsigned long long ga = (unsigned long long)(uintptr_t)gt;
    const u32 lds_off = (u32)(uintptr_t)&lds_gt[0];   // addrspace(3) byte offset
    u32x4 g0;
    g0.x = 1u;                                        // count=1, user descriptor
    g0.y = lds_off;                                   // lds_addr [63:32]
    g0.z = (u32)(ga & 0xffffffffu);                   // global_addr [95:64]
    g0.w = (u32)((ga >> 32) & 0x01ffffffu)            // global_addr [120:96]
         | (2u << 30);                                // type=2 ("image")
    i32x8 g1;
    g1[0] = 0x00030000;        // workgroup_mask=0, data_size=3 (8-byte units)
    g1[1] = (int)(2048u << 16);// tensor_dim0 = 2048 (lo16 at bits 63:48)
    g1[2] = (int)(1u << 16);   // tensor_dim0 hi=0; tensor_dim1 = 1 (lo16)
    g1[3] = (int)(2048u << 16);// tensor_dim1 hi=0; tile_dim0 = 2048
    g1[4] = 1;                 // tile_dim1 = 1; tile_dim2 = 0
    g1[5] = 2048;              // tensor_dim0_stride[31:0] = 2048 (8B units)
    g1[6] = 0;                 // stride0 hi, dim1_stride lo
    g1[7] = 0;
    i32x4 z4 = {0, 0, 0, 0};
    i32x8 z8 = {0, 0, 0, 0, 0, 0, 0, 0};
    __builtin_amdgcn_tensor_load_to_lds(g0, g1, z4, z4, z8, 0);
    __builtin_amdgcn_s_wait_tensorcnt(0);
  }
#else
  // Fallback: cooperative vector fill.
  for (int i = threadIdx.x; i < GT_FLOATS / 4; i += 256)
    ((float4*)lds_gt)[i] = ((const float4*)gt)[i];
#endif
  __syncthreads();

  // ---- Build B matrix (4x16, K x N) once: column j = [-2yx, -2yy, 1, |y_j|^2]
  // B layout: lanes 0-15 hold K0,K1; lanes 16-31 hold K2,K3 (col = lane&15).
  const float BIG = 1e30f;
  const int jc = valid ? j : (MAXN - 1);          // clamp, branch-free load
  const float yx = y[b * (MAXN * 2) + jc * 2 + 0];
  const float yy = y[b * (MAXN * 2) + jc * 2 + 1];
  const float k0 = valid ? (-2.0f * yx) : 0.0f;   // padded cols: d2 = |g|^2+BIG
  const float k1 = valid ? (-2.0f * yy) : 0.0f;
  const float k3 = valid ? (yx * yx + yy * yy) : BIG;
  v2f bm;
  bm.x = hi ? 1.0f : k0;
  bm.y = hi ? k3 : k1;

  float best = BIG;

  // Each wave: 16 tiles of 16 gt-rows => rotation candidates g in [32w, 32w+32).
  const int t0 = wave * 16;
#pragma unroll 4
  for (int t = t0; t < t0 + 16; ++t) {
    const int row = t * 16 + j;                   // gt row in [0, 2048)
    const float px = lds_gt[row * 2 + 0];         // ds_load_b64 per lane
    const float py = lds_gt[row * 2 + 1];
    // A (16x4) layout: lanes 0-15 hold K0,K1 = (gx,gy) for row M=lane;
    //                  lanes 16-31 hold K2,K3 = (|g|^2, 1) for row M=lane-16.
    v2f am;
    am.x = hi ? (px * px + py * py) : px;
    am.y = hi ? 1.0f : py;

    v8f c = {0.f, 0.f, 0.f, 0.f, 0.f, 0.f, 0.f, 0.f};
    // D[m][n] = ||g_m - y_n||^2  (one full 16x16 distance tile per WMMA)
    v8f d = __builtin_amdgcn_wmma_f32_16x16x4_f32(
        false, am, false, bm, (short)0, c, false, false);

    // D layout: lane = column j; VGPR r = row M=r (low half -> group 2t)
    // or M=r+8 (high half -> group 2t+1). Each half reduces its own group.
    // Valid columns are lanes 0..7 of each half; pads hold +BIG distances.

    // (a) per-lane column min over the 8 rows (min_i for column j)
    float cmin = fminf(fminf(fminf(d[0], d[1]), fminf(d[2], d[3])),
                       fminf(fminf(d[4], d[5]), fminf(d[6], d[7])));

    // (b) packed butterfly row-min: 8 rows x 8 valid lanes in 8+4+2 shuffles.
    float m0 = fminf(d[0], __shfl_xor(d[0], 4, 32));
    float m1 = fminf(d[1], __shfl_xor(d[1], 4, 32));
    float m2 = fminf(d[2], __shfl_xor(d[2], 4, 32));
    float m3 = fminf(d[3], __shfl_xor(d[3], 4, 32));
    float m4 = fminf(d[4], __shfl_xor(d[4], 4, 32));
    float m5 = fminf(d[5], __shfl_xor(d[5], 4, 32));
    float m6 = fminf(d[6], __shfl_xor(d[6], 4, 32));
    float m7 = fminf(d[7], __shfl_xor(d[7], 4, 32));
    const bool p4 = (lane & 4) != 0;
    float u0 = p4 ? m1 : m0;                      // rows packed 2-per-register
    float u1 = p4 ? m3 : m2;
    float u2 = p4 ? m5 : m4;
    float u3 = p4 ? m7 : m6;
    u0 = fminf(u0, __shfl_xor(u0, 2, 32));
    u1 = fminf(u1, __shfl_xor(u1, 2, 32));
    u2 = fminf(u2, __shfl_xor(u2, 2, 32));
    u3 = fminf(u3, __shfl_xor(u3, 2, 32));
    const bool p2 = (lane & 2) != 0;
    float w0 = p2 ? u1 : u0;                      // 4 rows per register
    float w1 = p2 ? u3 : u2;
    w0 = fminf(w0, __shfl_xor(w0, 1, 32));
    w1 = fminf(w1, __shfl_xor(w1, 1, 32));
    float z = (lane & 1) ? w1 : w0;               // lanes 0..7: all 8 row-mins

    // (c) one shared 3-step add butterfly sums row-mins AND column-mins:
    //     lanes 0..7 are exactly the 8 valid columns, so fold cmin in now.
    float s = z + cmin;
    s += __shfl_xor(s, 1, 32);
    s += __shfl_xor(s, 2, 32);
    s += __shfl_xor(s, 4, 32);                    // = sum_i min_j + sum_j min_i

    best = fminf(best, s * 0.125f);               // valid in lanes 0..7 / 16..23
  }

  // low half holds best over even groups, high half over odd groups
  best = fminf(best, __shfl_xor(best, 16, 32));

  if (lane == 0) sbest[wave] = best;
  __syncthreads();
  if (threadIdx.x == 0) {
    float v = sbest[0];
#pragma unroll
    for (int w = 1; w < 8; ++w) v = fminf(v, sbest[w]);
    out[b] = v;
  }
}

extern "C" void kernel_launch(void* const* d_in, const int* in_sizes, int n_in,
                              void* d_out, int out_size, void* d_ws, size_t ws_size,
                              hipStream_t stream) {
  (void)n_in; (void)out_size; (void)d_ws; (void)ws_size;
  const float* y      = (const float*)d_in[0];   // [B,8,2] f32
  const int*   n      = (const int*)d_in[1];     // [B] i32
  const float* labels = (const float*)d_in[2];   // [6] f32
  const float* ex     = (const float*)d_in[3];   // [6,256,8,2] f32
  float* out = (float*)d_out;                    // [B] f32
  const int B = in_sizes[1];                     // 4096
  rotpoly_chamfer_wmma<<<B, 256, 0, stream>>>(y, n, labels, ex, out, B);
}